// AttributeDecoder_77043123356187
// MI455X (gfx1250) — compile-verified
//
#include <hip/hip_runtime.h>

typedef __attribute__((ext_vector_type(2))) float v2f;
typedef __attribute__((ext_vector_type(8))) float v8f;

#define TPB 256

__global__ void fill_ones_kernel(float* __restrict__ p, int n) {
  int i = blockIdx.x * blockDim.x + threadIdx.x;
  if (i < n) p[i] = 1.0f;
}

__global__ void degree_kernel(const int* __restrict__ dst, float* __restrict__ deg, int E) {
  int e = blockIdx.x * blockDim.x + threadIdx.x;
  if (e < E) unsafeAtomicAdd(&deg[dst[e]], 1.0f);
}

__global__ void rsqrt_kernel(float* __restrict__ p, int n) {
  int i = blockIdx.x * blockDim.x + threadIdx.x;
  if (i < n) {
    float d = p[i];
    p[i] = d > 0.0f ? rsqrtf(d) : 0.0f;
  }
}

// C[M,N] = A[M,K] @ B[K,N], all row-major fp32, via V_WMMA_F32_16X16X4_F32.
// One wave computes one 16x16 output tile; K consumed 4 at a time.
// Requires M%16==0, N%16==0, K%4==0. Tile guard is wave-uniform so EXEC
// is all-ones for every wave that reaches the WMMA (ISA requirement).
__global__ void gemm_wmma_kernel(const float* __restrict__ A, const float* __restrict__ B,
                                 float* __restrict__ C, int M, int N, int K) {
  int lane = threadIdx.x & 31;
  int wave = (blockIdx.x * blockDim.x + threadIdx.x) >> 5;
  int tilesN = N >> 4;
  int numTiles = (M >> 4) * tilesN;
  if (wave >= numTiles) return;
  int tM = wave / tilesN;
  int tN = wave - tM * tilesN;
  int mr = lane & 15;          // row within A tile / col within B,C tile
  int kh = (lane >> 4) << 1;   // lanes 0-15: K offset 0,1 ; lanes 16-31: K offset 2,3

  const float* Arow = A + (size_t)((tM << 4) + mr) * K;
  const float* Bcol = B + (tN << 4) + mr;

  v8f acc = {};
  for (int k = 0; k < K; k += 4) {
    v2f a, b;
    a.x = Arow[k + kh + 0];
    a.y = Arow[k + kh + 1];
    b.x = Bcol[(size_t)(k + kh + 0) * N];
    b.y = Bcol[(size_t)(k + kh + 1) * N];
    acc = __builtin_amdgcn_wmma_f32_16x16x4_f32(false, a, false, b, (short)0, acc,
                                                false, false);
  }
  // C/D layout: VGPR r -> row (tM*16 + r + 8*(lane>=16)), col tN*16 + (lane&15)
  float* Crow = C + (size_t)((tM << 4) + ((lane >> 4) << 3)) * N + (tN << 4) + mr;
#pragma unroll
  for (int r = 0; r < 8; ++r) Crow[(size_t)r * N] = acc[r];
}

// out[i,:] = h[i,:] * dinv[i]^2  (self-loop contribution + accumulator init)
// processed in float4 chunks; f4Shift = log2(F/4)
__global__ void selfloop_init_kernel(const float* __restrict__ h, const float* __restrict__ dinv,
                                     float* __restrict__ out, long long total, int f4Shift) {
  long long i = (long long)blockIdx.x * blockDim.x + threadIdx.x;
  if (i >= total) return;
  int node = (int)(i >> f4Shift);
  float di = dinv[node];
  float s = di * di;
  float4 v = ((const float4*)h)[i];
  v.x *= s; v.y *= s; v.z *= s; v.w *= s;
  ((float4*)out)[i] = v;
}

// per-edge scatter: out[dst,:] += h[src,:] * dinv[src]*dinv[dst]
// (1<<cShift) threads per edge, each handling a float4 feature chunk.
__global__ void scatter_kernel(const float* __restrict__ h, const int* __restrict__ src,
                               const int* __restrict__ dst, const float* __restrict__ dinv,
                               float* __restrict__ out, int E, int F, int cShift) {
  long long tid = (long long)blockIdx.x * blockDim.x + threadIdx.x;
  int e = (int)(tid >> cShift);
  if (e >= E) return;
  int c = (int)(tid & ((1 << cShift) - 1));
  int s = src[e];
  int d = dst[e];
  float norm = dinv[s] * dinv[d];
  float4 v = *((const float4*)(h + (size_t)s * F) + c);
  float* op = out + (size_t)d * F + ((size_t)c << 2);
  unsafeAtomicAdd(op + 0, v.x * norm);
  unsafeAtomicAdd(op + 1, v.y * norm);
  unsafeAtomicAdd(op + 2, v.z * norm);
  unsafeAtomicAdd(op + 3, v.w * norm);
}

__global__ void bias_relu_kernel(float* __restrict__ x, const float* __restrict__ b,
                                 long long total, int fmask) {
  long long i = (long long)blockIdx.x * blockDim.x + threadIdx.x;
  if (i >= total) return;
  int f = (int)(i & fmask);
  x[i] = fmaxf(x[i] + b[f], 0.0f);
}

extern "C" void kernel_launch(void* const* d_in, const int* in_sizes, int n_in,
                              void* d_out, int out_size, void* d_ws, size_t ws_size,
                              hipStream_t stream) {
  const float* z  = (const float*)d_in[0];
  const int*   ei = (const int*)d_in[1];
  const float* W1 = (const float*)d_in[2];
  const float* b1 = (const float*)d_in[3];
  const float* W2 = (const float*)d_in[4];
  const float* b2 = (const float*)d_in[5];

  const int hidden = in_sizes[3];            // 128
  const int outch  = in_sizes[5];            // 64
  const int latent = in_sizes[2] / hidden;   // 64
  const int Nn     = in_sizes[0] / latent;   // 50000
  const int E      = in_sizes[1] / 2;        // 800000
  const int* src = ei;
  const int* dst = ei + E;

  // workspace layout: [dinv | bufA (h1 then h2) | bufB (x)]
  char* ws = (char*)d_ws;
  size_t off = 0;
  float* dinv = (float*)(ws + off); off += ((size_t)Nn * 4 + 255) & ~(size_t)255;
  float* bufA = (float*)(ws + off); off += ((size_t)Nn * hidden * 4 + 255) & ~(size_t)255;
  float* bufB = (float*)(ws + off);
  float* out  = (float*)d_out;

  auto blocks = [](long long n) { return (int)((n + TPB - 1) / TPB); };

  // shifts for power-of-two feature counts
  int h4shift = (hidden == 128) ? 5 : (hidden == 64 ? 4 : 31 - __builtin_clz(hidden / 4));
  int o4shift = (outch  == 128) ? 5 : (outch  == 64 ? 4 : 31 - __builtin_clz(outch  / 4));

  // degree (incl. self-loop) then dinv = rsqrt(deg)
  fill_ones_kernel<<<blocks(Nn), TPB, 0, stream>>>(dinv, Nn);
  degree_kernel<<<blocks(E), TPB, 0, stream>>>(dst, dinv, E);
  rsqrt_kernel<<<blocks(Nn), TPB, 0, stream>>>(dinv, Nn);

  // ---- layer 1 ----
  gemm_wmma_kernel<<<blocks((long long)(Nn / 16) * (hidden / 16) * 32), TPB, 0, stream>>>(
      z, W1, bufA, Nn, hidden, latent);
  selfloop_init_kernel<<<blocks((long long)Nn * (hidden / 4)), TPB, 0, stream>>>(
      bufA, dinv, bufB, (long long)Nn * (hidden / 4), h4shift);
  scatter_kernel<<<blocks((long long)E * (hidden / 4)), TPB, 0, stream>>>(
      bufA, src, dst, dinv, bufB, E, hidden, h4shift);
  bias_relu_kernel<<<blocks((long long)Nn * hidden), TPB, 0, stream>>>(
      bufB, b1, (long long)Nn * hidden, hidden - 1);

  // ---- layer 2 ----
  gemm_wmma_kernel<<<blocks((long long)(Nn / 16) * (outch / 16) * 32), TPB, 0, stream>>>(
      bufB, W2, bufA, Nn, outch, hidden);
  selfloop_init_kernel<<<blocks((long long)Nn * (outch / 4)), TPB, 0, stream>>>(
      bufA, dinv, out, (long long)Nn * (outch / 4), o4shift);
  scatter_kernel<<<blocks((long long)E * (outch / 4)), TPB, 0, stream>>>(
      bufA, src, dst, dinv, out, E, outch, o4shift);
  bias_relu_kernel<<<blocks((long long)Nn * outch), TPB, 0, stream>>>(
      out, b2, (long long)Nn * outch, outch - 1);
}